// RandomSortCM_15796889715208
// MI455X (gfx1250) — compile-verified
//
#include <hip/hip_runtime.h>
#include <hip/hip_bf16.h>

typedef __attribute__((ext_vector_type(2))) float v2f;
typedef __attribute__((ext_vector_type(8))) float v8f;

#define NDIM 256
#define NPAIRS 32896  // N*(N+1)/2

// One workgroup per batch sample. 256 threads = 8 wave32.
__global__ __launch_bounds__(256) void RandomSortCM_kernel(
    const float* __restrict__ X, const float* __restrict__ eps,
    const int* __restrict__ noise, float* __restrict__ out) {
  __shared__ float ssum[NDIM];   // row sum-of-squares
  __shared__ float skey[NDIM];   // sort keys
  __shared__ int   sidx[NDIM];   // permutation indices
  __shared__ float g[NDIM];      // gathered row-0 of permuted matrix

  const int b    = blockIdx.x;
  const int tid  = threadIdx.x;
  const int lane = tid & 31;
  const int wave = tid >> 5;
  const float* Xb = X + (size_t)b * NDIM * NDIM;

  // ---- Phase 1: row sums of squares via V_WMMA_F32_16X16X4_F32, B = ones ----
  // A layout (16x4 f32): lanes 0-15 hold row M=lane, K = {k0, k0+1};
  //                      lanes 16-31 hold row M=lane-16, K = {k0+2, k0+3}.
  // With B all-ones, D[m][n] = sum_k A[m][k] for every n.
  {
    const int mrow = lane & 15;
    const int koff = (lane >> 4) << 1;  // 0 or 2
    const v2f bones = {1.0f, 1.0f};
    for (int grp = wave * 2; grp < wave * 2 + 2; ++grp) {
      const int r0 = grp * 16;
      const float2* rp2 =
          (const float2*)(Xb + (size_t)(r0 + mrow) * NDIM + koff);
      v8f acc = {};
      for (int k0 = 0; k0 < NDIM; k0 += 4) {
        float2 xv = rp2[k0 >> 1];                 // global_load_b64
        v2f a = {xv.x * xv.x, xv.y * xv.y};       // square the elements
        acc = __builtin_amdgcn_wmma_f32_16x16x4_f32(
            false, a, false, bones, (short)0, acc, false, false);
      }
      // D layout: VGPR j -> M=j (lanes 0-15) / M=j+8 (lanes 16-31); every
      // lane of a half holds the same value (all N columns equal).
      if (lane == 0) {
#pragma unroll
        for (int j = 0; j < 8; ++j) ssum[r0 + j] = acc[j];
      } else if (lane == 16) {
#pragma unroll
        for (int j = 0; j < 8; ++j) ssum[r0 + 8 + j] = acc[j];
      }
    }
  }
  __syncthreads();

  // ---- Phase 2: keys + stable ascending bitonic argsort (index tiebreak) ----
  {
    const float noisef = (float)noise[0];
    skey[tid] = -sqrtf(ssum[tid]) + eps[(size_t)b * NDIM + tid] * noisef;
    sidx[tid] = tid;
  }
  __syncthreads();
  for (int k = 2; k <= NDIM; k <<= 1) {
    for (int j = k >> 1; j > 0; j >>= 1) {
      const int ixj = tid ^ j;
      if (ixj > tid) {
        const float ka = skey[tid], kb = skey[ixj];
        const int   ia = sidx[tid], ib = sidx[ixj];
        // strict lexicographic (key, idx) ordering; idx tiebreak == stable
        const bool cmp = (ka > kb) || (ka == kb && ia > ib);
        const bool asc = ((tid & k) == 0);
        if (cmp == asc) {
          skey[tid] = kb; skey[ixj] = ka;
          sidx[tid] = ib; sidx[ixj] = ia;
        }
      }
      __syncthreads();
    }
  }

  // ---- Phase 3: gather g[c] = X[b, inds[0], inds[c]] ----
  {
    const int ind0 = sidx[0];
    g[tid] = Xb[(size_t)ind0 * NDIM + sidx[tid]];
  }
  __syncthreads();

  // ---- Phase 4: triu-pair expansion, coalesced float2 stores ----
  // For row r: pairs j=0..(N-r-1): out[2*(s_r+j)] = g[r], out[2*(s_r+j)+1] = g[r+j]
  {
    float2* outp = (float2*)(out + (size_t)b * (2 * (size_t)NPAIRS));
    for (int r = 0; r < NDIM; ++r) {
      const int L = NDIM - r;
      const int s = r * NDIM - (r * (r - 1)) / 2;
      const float gr = g[r];
      for (int j = tid; j < L; j += 256) {
        float2 v;
        v.x = gr;
        v.y = g[r + j];
        outp[s + j] = v;                           // global_store_b64
      }
    }
  }
}

extern "C" void kernel_launch(void* const* d_in, const int* in_sizes, int n_in,
                              void* d_out, int out_size, void* d_ws, size_t ws_size,
                              hipStream_t stream) {
  const float* X     = (const float*)d_in[0];
  const float* eps   = (const float*)d_in[1];
  const int*   noise = (const int*)d_in[2];
  float*       out   = (float*)d_out;
  const int B = in_sizes[0] / (NDIM * NDIM);
  hipLaunchKernelGGL(RandomSortCM_kernel, dim3(B), dim3(256), 0, stream,
                     X, eps, noise, out);
}